// MetaFoam_45019847197262
// MI455X (gfx1250) — compile-verified
//
#include <hip/hip_runtime.h>
#include <math.h>

#define VOCAB   32768
#define DDIM    512
#define NB      3
#define NF      3
#define SEQ     4
#define NU      9
#define COUPLING 0.3f
#define DIVW    2.0f
#define MAXENT  10.397207708399179f   // log(32768)

#define MROWS   128      // vocab rows per GEMM block
#define KC      32       // K chunk (panel width)
#define NCH     (DDIM / KC)
#define LDP     33       // padded LDS row stride in dwords (bank-conflict-free, odd)

typedef __attribute__((ext_vector_type(2))) float v2f;
typedef __attribute__((ext_vector_type(8))) float v8f;
typedef __attribute__((ext_vector_type(4))) unsigned int u32x4;
typedef __attribute__((ext_vector_type(4))) int i32x4;
typedef __attribute__((ext_vector_type(8))) int i32x8;

#if defined(__HIP_DEVICE_COMPILE__) && __has_builtin(__builtin_amdgcn_tensor_load_to_lds) && __has_builtin(__builtin_amdgcn_s_wait_tensorcnt)
#define USE_TDM 1
#else
#define USE_TDM 0
#endif

// ---- workspace layout (floats) ----
#define WS_MEM   0                     // [NF*NB*DDIM] = 4608 scan carry
#define WS_UPAD  4608                  // [16*DDIM] normalized states, rows 9..15 zero
#define WS_WBUB  12800                 // [9] bubble weights
#define WS_BORN  12816                 // [NF*VOCAB] per-foam logits
#define WS_STAT  (12816 + NF*VOCAB)    // [32] scalars
// stats indices
#define ST_MAX   0   // [3]
#define ST_Z     3   // [3]
#define ST_W     6   // [3] meta weights
#define ST_CJ    9   // [9] weights_k * wbub
#define ST_MAXM  18
#define ST_ZM    19

// ---- output layout (floats) ----
#define O_TOK   0
#define O_RHO   (SEQ*VOCAB)
#define O_W     (O_RHO + SEQ*DDIM*DDIM)
#define O_S     (O_W + SEQ*NF)
#define O_H     (O_S + SEQ)
#define O_F     (O_H + SEQ)
#define O_DIST  (O_F + SEQ)

__device__ __forceinline__ float blockReduceSum(float v, float* red) {
  int tid = threadIdx.x;
  red[tid] = v; __syncthreads();
  for (int s = blockDim.x >> 1; s > 0; s >>= 1) {
    if (tid < s) red[tid] += red[tid + s];
    __syncthreads();
  }
  float r = red[0]; __syncthreads();
  return r;
}

__device__ __forceinline__ float blockReduceMax(float v, float* red) {
  int tid = threadIdx.x;
  red[tid] = v; __syncthreads();
  for (int s = blockDim.x >> 1; s > 0; s >>= 1) {
    if (tid < s) red[tid] = fmaxf(red[tid], red[tid + s]);
    __syncthreads();
  }
  float r = red[0]; __syncthreads();
  return r;
}

#if USE_TDM
// Issue a TDM 2D-tile load: E panel [MROWS x KC] f32, row stride DDIM elements,
// into LDS at ldsOff with +1-dword padding after every KC dwords (stride = LDP).
__device__ __forceinline__ void tdm_load_panel(const float* gsrc, unsigned ldsOff) {
  unsigned long long ga = (unsigned long long)(uintptr_t)gsrc;
  u32x4 g0 = { 1u,                       // count=1, user descriptor
               ldsOff,                   // lds_addr (bytes)
               (unsigned)(ga & 0xffffffffu),
               (unsigned)((ga >> 32) & 0x01ffffffu) | (2u << 30) };  // type=2 (image)
  // group1: data_size=4B(code2), pad_enable, pad_interval code4 (=32 dwords), pad_amount 0 (=1 dword)
  i32x8 g1 = { (int)((2u << 16) | (1u << 20) | (4u << 22)),   // d0
               (int)((unsigned)KC << 16),                     // tensor_dim0[15:0] in d1[31:16]
               (int)((unsigned)MROWS << 16),                  // tensor_dim1[15:0] in d2[31:16]
               (int)((unsigned)KC << 16),                     // tile_dim0 in d3[31:16]
               (int)MROWS,                                    // tile_dim1 in d4[15:0], tile_dim2=0
               (int)DDIM,                                     // tensor_dim0_stride low32
               0, 0 };
  i32x4 g2 = { 0, 0, 0, 0 };
  i32x4 g3 = { 0, 0, 0, 0 };
#if __clang_major__ >= 23
  __builtin_amdgcn_tensor_load_to_lds(g0, g1, g2, g3, (i32x8)0, 0);
#else
  __builtin_amdgcn_tensor_load_to_lds(g0, g1, g2, g3, 0);
#endif
}
#endif

// ---------------- Kernel A: per-foam step (memory, novelty, states, eq, normalize) ----------------
__global__ __launch_bounds__(256)
void foam_kernel(const int* tokens, int t, const float* E, const float* Ws,
                 const float* p_dbase, const float* p_sens,
                 float* mem, float* Upad, float* wbub) {
  __shared__ float red[256];
  __shared__ float xp[DDIM];
  const int f = blockIdx.x;
  const int tid = threadIdx.x;
  const int tok = tokens[t];
  const float dbase = *p_dbase;
  const float sens  = fabsf(*p_sens);
  const float* x = E + (size_t)tok * DDIM;
  float* memf = mem + (size_t)f * NB * DDIM;

  const int d0 = tid, d1 = tid + 256;
  float x0 = x[d0], x1 = x[d1];
  float mm0 = (memf[d0] + memf[DDIM + d0] + memf[2*DDIM + d0]) * (1.0f/3.0f);
  float mm1 = (memf[d1] + memf[DDIM + d1] + memf[2*DDIM + d1]) * (1.0f/3.0f);

  float sx = blockReduceSum(x0*x0 + x1*x1, red);
  float sm = blockReduceSum(mm0*mm0 + mm1*mm1, red);
  float sd = blockReduceSum(x0*mm0 + x1*mm1, red);

  float mem_norm = sqrtf(sm) + 1e-10f;
  float x_norm   = sqrtf(sx) + 1e-10f;
  float novelty  = (mem_norm > 1e-8f) ? (1.0f - sd / (x_norm * mem_norm)) : 1.0f;
  float decay    = 1.0f / (1.0f + expf(-(dbase - sens * novelty)));

  xp[d0] = x0 + decay * mm0;
  xp[d1] = x1 + decay * mm1;
  __syncthreads();

  float s[NB][2];
  for (int n = 0; n < NB; ++n) {
    const float* Wn = Ws + (((size_t)f * NB + n) * DDIM) * DDIM;
    float a0 = 0.0f, a1 = 0.0f;
    #pragma unroll 4
    for (int d = 0; d < DDIM; ++d) {
      float xv = xp[d];
      a0 = fmaf(xv, Wn[(size_t)d * DDIM + d0], a0);
      a1 = fmaf(xv, Wn[(size_t)d * DDIM + d1], a1);
    }
    s[n][0] = a0; s[n][1] = a1;
  }
  #pragma unroll
  for (int it = 0; it < 3; ++it) {
    float m0 = (s[0][0] + s[1][0] + s[2][0]) * (1.0f/3.0f);
    float m1 = (s[0][1] + s[1][1] + s[2][1]) * (1.0f/3.0f);
    #pragma unroll
    for (int n = 0; n < NB; ++n) {
      s[n][0] += COUPLING * (m0 - s[n][0]);
      s[n][1] += COUPLING * (m1 - s[n][1]);
    }
  }
  float nrm[NB];
  for (int n = 0; n < NB; ++n)
    nrm[n] = sqrtf(blockReduceSum(s[n][0]*s[n][0] + s[n][1]*s[n][1], red));
  float mx = fmaxf(fmaxf(nrm[0], nrm[1]), nrm[2]) * DIVW;
  float e0 = expf(DIVW*nrm[0] - mx), e1 = expf(DIVW*nrm[1] - mx), e2 = expf(DIVW*nrm[2] - mx);
  float inv = 1.0f / (e0 + e1 + e2);
  if (tid == 0) {
    wbub[f*NB + 0] = e0 * inv;
    wbub[f*NB + 1] = e1 * inv;
    wbub[f*NB + 2] = e2 * inv;
  }
  for (int n = 0; n < NB; ++n) {
    float r = 1.0f / (nrm[n] + 1e-10f);
    Upad[(size_t)(f*NB + n) * DDIM + d0] = s[n][0] * r;
    Upad[(size_t)(f*NB + n) * DDIM + d1] = s[n][1] * r;
    float* mr = memf + (size_t)n * DDIM;
    mr[d0] = decay * mr[d0] + (1.0f - decay) * s[n][0];
    mr[d1] = decay * mr[d1] + (1.0f - decay) * s[n][1];
  }
}

// ---------------- Kernel B: P = E @ Upad^T (f32 WMMA), TDM double-buffered staging ----------------
__global__ __launch_bounds__(256)
void born_gemm_kernel(const float* E, const float* Upad, const float* wbub, float* born) {
  __shared__ float Et[2][MROWS * LDP];   // E panels (TDM pads rows to LDP dwords)
  __shared__ float Ut[2][16 * LDP];      // U panels
  __shared__ float wbs[16];
  const int tid  = threadIdx.x;
  const int wave = tid >> 5;
  const int lane = tid & 31;
  const size_t rowBase = (size_t)blockIdx.x * MROWS;

  if (tid < 16) wbs[tid] = (tid < NU) ? wbub[tid] : 0.0f;

  v8f acc = {0.f,0.f,0.f,0.f,0.f,0.f,0.f,0.f};
  const int m  = lane & 15;
  const int kb = (lane >> 4) * 2;
  const int nc = lane & 15;

#if USE_TDM
  if (wave == 0)   // one DMA per panel, issued by wave 0, tracked by TENSORcnt
    tdm_load_panel(E + rowBase * DDIM, (unsigned)(uintptr_t)(const void*)&Et[0][0]);
#endif

  for (int c = 0; c < NCH; ++c) {
    const int buf = c & 1;
    const int k0 = c * KC;
#if USE_TDM
    if (wave == 0 && c + 1 < NCH)
      tdm_load_panel(E + rowBase * DDIM + (k0 + KC),
                     (unsigned)(uintptr_t)(const void*)&Et[buf ^ 1][0]);
    { // stage U panel [16 x KC] synchronously (tiny, L2-resident)
      int r = tid >> 4, cc = (tid & 15) * 2;
      const float2 v = *(const float2*)(Upad + (size_t)r * DDIM + k0 + cc);
      Ut[buf][r * LDP + cc]     = v.x;
      Ut[buf][r * LDP + cc + 1] = v.y;
    }
    if (wave == 0) {   // immediate must be a compile-time constant
      if (c + 1 < NCH) __builtin_amdgcn_s_wait_tensorcnt(1);
      else             __builtin_amdgcn_s_wait_tensorcnt(0);
    }
    __syncthreads();
#else
    for (int i = tid; i < MROWS * (KC/4); i += 256) {
      int r = i / (KC/4), c4 = i % (KC/4);
      const float4 v = *(const float4*)(E + (rowBase + r) * DDIM + k0 + c4*4);
      float* dst = &Et[buf][r * LDP + c4*4];
      dst[0] = v.x; dst[1] = v.y; dst[2] = v.z; dst[3] = v.w;
    }
    {
      int r = tid >> 4, cc = (tid & 15) * 2;
      const float2 v = *(const float2*)(Upad + (size_t)r * DDIM + k0 + cc);
      Ut[buf][r * LDP + cc]     = v.x;
      Ut[buf][r * LDP + cc + 1] = v.y;
    }
    if (k0 + KC < DDIM)
      __builtin_prefetch(E + (rowBase + (tid >> 1)) * DDIM + k0 + KC + (tid & 1) * 16, 0, 0);
    __syncthreads();
#endif

    const float* Erow = &Et[buf][(wave * 16 + m) * LDP];
    const float* Urow = &Ut[buf][nc * LDP];
    for (int kk = 0; kk < KC; kk += 4) {
      v2f a = { Erow[kk + kb], Erow[kk + kb + 1] };
      v2f b = { Urow[kk + kb], Urow[kk + kb + 1] };
      acc = __builtin_amdgcn_wmma_f32_16x16x4_f32(false, a, false, b,
                                                  (short)0, acc, false, false);
    }
    __syncthreads();   // panel buffer may be overwritten next iteration
  }

  // lane holds P for (row = wave*16 + i + 8*(lane>>4), col = nc); combine triples per foam
  const float wb = wbs[nc];
  #pragma unroll
  for (int i = 0; i < 8; ++i) {
    int mL = i + 8 * (lane >> 4);
    float p = acc[i];
    Et[0][(wave * 16 + mL) * 16 + nc] = wb * p * p;   // reuse Et[0] as [128][16]
  }
  __syncthreads();
  if (tid < MROWS) {
    #pragma unroll
    for (int k = 0; k < NF; ++k) {
      float s = Et[0][tid*16 + k*3] + Et[0][tid*16 + k*3 + 1] + Et[0][tid*16 + k*3 + 2];
      born[(size_t)k * VOCAB + rowBase + tid] = s;
    }
  }
}

// ---------------- Kernel C: softmax stats, agreement, meta weights, meta softmax stats ----------
__global__ __launch_bounds__(1024)
void meta_stats_kernel(const float* born, const float* wbub, const float* p_temp,
                       float* stats, float* outW, float* outH) {
  __shared__ float red[1024];
  __shared__ float bc[4];
  const int tid = threadIdx.x;
  const float* b0 = born;
  const float* b1 = born + VOCAB;
  const float* b2 = born + 2*VOCAB;

  float m0 = -3.4e38f, m1 = -3.4e38f, m2 = -3.4e38f;
  for (int v = tid; v < VOCAB; v += 1024) {
    m0 = fmaxf(m0, b0[v]); m1 = fmaxf(m1, b1[v]); m2 = fmaxf(m2, b2[v]);
  }
  m0 = blockReduceMax(m0, red); m1 = blockReduceMax(m1, red); m2 = blockReduceMax(m2, red);

  float z0=0,z1=0,z2=0, s0=0,s1=0,s2=0, q00=0,q11=0,q22=0, q01=0,q02=0,q12=0;
  for (int v = tid; v < VOCAB; v += 1024) {
    float d0v = b0[v]-m0, d1v = b1[v]-m1, d2v = b2[v]-m2;
    float e0 = expf(d0v), e1 = expf(d1v), e2 = expf(d2v);
    z0+=e0; z1+=e1; z2+=e2;
    s0 = fmaf(e0,d0v,s0); s1 = fmaf(e1,d1v,s1); s2 = fmaf(e2,d2v,s2);
    q00 = fmaf(e0,e0,q00); q11 = fmaf(e1,e1,q11); q22 = fmaf(e2,e2,q22);
    q01 = fmaf(e0,e1,q01); q02 = fmaf(e0,e2,q02); q12 = fmaf(e1,e2,q12);
  }
  z0=blockReduceSum(z0,red); z1=blockReduceSum(z1,red); z2=blockReduceSum(z2,red);
  s0=blockReduceSum(s0,red); s1=blockReduceSum(s1,red); s2=blockReduceSum(s2,red);
  q00=blockReduceSum(q00,red); q11=blockReduceSum(q11,red); q22=blockReduceSum(q22,red);
  q01=blockReduceSum(q01,red); q02=blockReduceSum(q02,red); q12=blockReduceSum(q12,red);

  if (tid == 0) {
    float Z[3]   = {z0,z1,z2};
    float S1[3]  = {s0,s1,s2};
    float MX[3]  = {m0,m1,m2};
    float Q[3][3]= {{q00,q01,q02},{q01,q11,q12},{q02,q12,q22}};
    float conc[3], nrm[3];
    for (int k = 0; k < 3; ++k) {
      float H = logf(Z[k]) - S1[k]/Z[k];
      conc[k] = 1.0f - H / MAXENT;
      nrm[k]  = sqrtf(Q[k][k]) / Z[k] + 1e-10f;
    }
    float A[3][3], agr[3];
    for (int k = 0; k < 3; ++k)
      for (int j = 0; j < 3; ++j)
        A[k][j] = (Q[k][j] / (Z[k]*Z[j])) / (nrm[k]*nrm[j]);
    for (int k = 0; k < 3; ++k)
      agr[k] = (A[k][0] + A[k][1] + A[k][2] - A[k][k]) * 0.5f;
    float temp = fmaxf(fabsf(*p_temp), 0.01f);
    float l0 = conc[0]*agr[0]/temp, l1 = conc[1]*agr[1]/temp, l2 = conc[2]*agr[2]/temp;
    float lm = fmaxf(fmaxf(l0,l1),l2);
    float w0 = expf(l0-lm), w1 = expf(l1-lm), w2 = expf(l2-lm);
    float wi = 1.0f/(w0+w1+w2);
    w0*=wi; w1*=wi; w2*=wi;
    float W[3] = {w0,w1,w2};
    for (int k = 0; k < 3; ++k) {
      stats[ST_MAX+k] = MX[k]; stats[ST_Z+k] = Z[k]; stats[ST_W+k] = W[k];
      outW[k] = W[k];
      for (int n = 0; n < 3; ++n) stats[ST_CJ + k*3 + n] = W[k] * wbub[k*3 + n];
    }
    bc[0]=w0; bc[1]=w1; bc[2]=w2;
  }
  __syncthreads();
  float w0 = bc[0], w1 = bc[1], w2 = bc[2];

  float mm = -3.4e38f;
  for (int v = tid; v < VOCAB; v += 1024)
    mm = fmaxf(mm, w0*b0[v] + w1*b1[v] + w2*b2[v]);
  mm = blockReduceMax(mm, red);
  float zm = 0.0f, sm = 0.0f;
  for (int v = tid; v < VOCAB; v += 1024) {
    float d = w0*b0[v] + w1*b1[v] + w2*b2[v] - mm;
    float e = expf(d);
    zm += e; sm = fmaf(e, d, sm);
  }
  zm = blockReduceSum(zm, red); sm = blockReduceSum(sm, red);
  if (tid == 0) {
    stats[ST_MAXM] = mm; stats[ST_ZM] = zm;
    *outH = logf(zm) - sm/zm;
  }
}

// ---------------- Kernel D: write token_probs and per-foam dists ----------------
__global__ __launch_bounds__(256)
void probs_kernel(const float* born, const float* stats, float* tokOut, float* distOut) {
  int v = blockIdx.x * 256 + threadIdx.x;
  float bm = 0.0f;
  #pragma unroll
  for (int k = 0; k < NF; ++k) {
    float b = born[(size_t)k * VOCAB + v];
    distOut[(size_t)k * VOCAB + v] = expf(b - stats[ST_MAX+k]) / stats[ST_Z+k];
    bm = fmaf(stats[ST_W+k], b, bm);
  }
  tokOut[v] = expf(bm - stats[ST_MAXM]) / stats[ST_ZM];
}

// ---------------- Kernel E: rho_meta = sum_j c_j u_j u_j^T ----------------
__global__ __launch_bounds__(256)
void rho_kernel(const float* Upad, const float* stats, float* rhoOut) {
  __shared__ float cj[NU];
  __shared__ float ui[NU];
  int idx = blockIdx.x * 256 + threadIdx.x;
  int i = idx >> 9, j = idx & (DDIM - 1);
  if (threadIdx.x < NU) {
    cj[threadIdx.x] = stats[ST_CJ + threadIdx.x];
    ui[threadIdx.x] = Upad[(size_t)threadIdx.x * DDIM + i];
  }
  __syncthreads();
  float s = 0.0f;
  #pragma unroll
  for (int l = 0; l < NU; ++l)
    s = fmaf(cj[l] * ui[l], Upad[(size_t)l * DDIM + j], s);
  rhoOut[idx] = s;
}

// ---------------- Kernel F: 9x9 Jacobi eig -> S_rho, F = H - S_rho ----------------
__global__ __launch_bounds__(128)
void eig_kernel(const float* Upad, const float* stats, const float* Hptr,
                float* sOut, float* fOut) {
  __shared__ float G[NU * NU];
  int tid = threadIdx.x;
  if (tid < NU * NU) {
    int l = tid / NU, mcol = tid % NU;
    float s = 0.0f;
    for (int d = 0; d < DDIM; ++d)
      s = fmaf(Upad[(size_t)l * DDIM + d], Upad[(size_t)mcol * DDIM + d], s);
    G[tid] = s;
  }
  __syncthreads();
  if (tid == 0) {
    float M[NU][NU], sc[NU];
    for (int l = 0; l < NU; ++l) sc[l] = sqrtf(fmaxf(stats[ST_CJ + l], 0.0f));
    for (int l = 0; l < NU; ++l)
      for (int m2 = 0; m2 < NU; ++m2)
        M[l][m2] = sc[l] * sc[m2] * G[l*NU + m2];
    for (int sweep = 0; sweep < 10; ++sweep) {
      for (int p = 0; p < NU - 1; ++p) {
        for (int q = p + 1; q < NU; ++q) {
          float apq = M[p][q];
          if (fabsf(apq) < 1e-18f) continue;
          float theta = 0.5f * atan2f(2.0f * apq, M[q][q] - M[p][p]);
          float c = cosf(theta), s = sinf(theta);
          for (int k = 0; k < NU; ++k) {
            float mpk = M[p][k], mqk = M[q][k];
            M[p][k] = c * mpk - s * mqk;
            M[q][k] = s * mpk + c * mqk;
          }
          for (int k = 0; k < NU; ++k) {
            float mkp = M[k][p], mkq = M[k][q];
            M[k][p] = c * mkp - s * mkq;
            M[k][q] = s * mkp + c * mkq;
          }
        }
      }
    }
    float ev[NU], sum = (float)(DDIM - NU) * 1e-12f;
    for (int l = 0; l < NU; ++l) { ev[l] = fmaxf(M[l][l], 1e-12f); sum += ev[l]; }
    float S = 0.0f;
    for (int l = 0; l < NU; ++l) {
      float p = ev[l] / sum;
      S -= p * fmaxf(logf(p), -100.0f);
    }
    float pt = 1e-12f / sum;
    S += (float)(DDIM - NU) * (-pt * fmaxf(logf(pt), -100.0f));
    *sOut = S;
    *fOut = *Hptr - S;
  }
}

extern "C" void kernel_launch(void* const* d_in, const int* in_sizes, int n_in,
                              void* d_out, int out_size, void* d_ws, size_t ws_size,
                              hipStream_t stream) {
  const int*   tokens = (const int*)d_in[0];
  const float* E      = (const float*)d_in[1];
  const float* Ws     = (const float*)d_in[2];
  const float* mtemp  = (const float*)d_in[3];
  const float* dbase  = (const float*)d_in[4];
  const float* sens   = (const float*)d_in[5];
  float* out = (float*)d_out;
  float* ws  = (float*)d_ws;

  float* mem   = ws + WS_MEM;
  float* Upad  = ws + WS_UPAD;
  float* wbub  = ws + WS_WBUB;
  float* born  = ws + WS_BORN;
  float* stats = ws + WS_STAT;

  // zero scan carry + Upad (incl. zero pad rows 9..15) each call; captured as a graph node
  (void)hipMemsetAsync(ws, 0, (size_t)WS_BORN * sizeof(float), stream);

  for (int t = 0; t < SEQ; ++t) {
    foam_kernel<<<NF, 256, 0, stream>>>(tokens, t, E, Ws, dbase, sens, mem, Upad, wbub);
    born_gemm_kernel<<<VOCAB / MROWS, 256, 0, stream>>>(E, Upad, wbub, born);
    meta_stats_kernel<<<1, 1024, 0, stream>>>(born, wbub, mtemp, stats,
                                              out + O_W + (size_t)t * NF,
                                              out + O_H + t);
    probs_kernel<<<VOCAB / 256, 256, 0, stream>>>(born, stats,
                                                  out + O_TOK + (size_t)t * VOCAB,
                                                  out + O_DIST + (size_t)t * NF * VOCAB);
    rho_kernel<<<(DDIM * DDIM) / 256, 256, 0, stream>>>(Upad, stats,
                                                        out + O_RHO + (size_t)t * DDIM * DDIM);
    eig_kernel<<<1, 128, 0, stream>>>(Upad, stats, out + O_H + t,
                                      out + O_S + t, out + O_F + t);
  }
}